// SE4_55224689492533
// MI455X (gfx1250) — compile-verified
//
#include <hip/hip_runtime.h>

typedef float v2f __attribute__((ext_vector_type(2)));
typedef float v4f __attribute__((ext_vector_type(4)));
typedef float v8f __attribute__((ext_vector_type(8)));

// Problem constants: B=8, C=64, H=W=256, SQ=256, 16C=1024.
// C*H*W = 2^22, H*W = 2^16.

// ---------------------------------------------------------------------------
// Kernel 1: per-patch mean pooling.
// One 256-thread block per (b, c, i, j) patch (8192 blocks). Each thread sums
// four float4 (16 elements); 64x64 patch = 4096 elements. Regular (RT) loads
// so t stays resident in the 192MB L2 for the gating pass.
// pooled layout matches reference cat order: pooled[b*1024 + (i*4+j)*64 + c].
// ---------------------------------------------------------------------------
__global__ __launch_bounds__(256) void se4_pool_kernel(
    const float* __restrict__ t, float* __restrict__ pooled) {
  int bid = blockIdx.x;
  int j = bid & 3;
  int i = (bid >> 2) & 3;
  int c = (bid >> 4) & 63;
  int b = bid >> 10;
  const float* base =
      t + ((size_t)(b * 64 + c) << 16) + i * (64 * 256) + j * 64;
  int tid = threadIdx.x;

  float sum = 0.0f;
#pragma unroll
  for (int k = 0; k < 4; ++k) {
    int f = tid + k * 256;          // 0..1023 float4 slots in the patch
    int row = f >> 4;               // 0..63
    int col4 = f & 15;              // 0..15
    v4f v = *(const v4f*)(base + row * 256 + col4 * 4);
    sum += v.x + v.y + v.z + v.w;
  }

  // wave32 tree reduction
#pragma unroll
  for (int off = 16; off > 0; off >>= 1)
    sum += __shfl_down(sum, off, 32);

  __shared__ float part[8];
  int lane = tid & 31;
  int wv = tid >> 5;
  if (lane == 0) part[wv] = sum;
  __syncthreads();
  if (tid == 0) {
    float s = 0.0f;
#pragma unroll
    for (int w = 0; w < 8; ++w) s += part[w];
    pooled[b * 1024 + ((i * 4 + j) << 6) + c] = s * (1.0f / 4096.0f);
  }
}

// ---------------------------------------------------------------------------
// Kernel 2: squeeze-excite MLP via V_WMMA_F32_16X16X4_F32.
// Single workgroup, 16 waves (512 threads). Batch M=8 padded to 16.
// Layer 1: S[16,256] = mish(P[16,1024] @ reduce_w^T + b); wave w owns N-tile w.
// Layer 2: G[16,1024] = sigmoid(S @ expand_w^T + b); wave w owns N-tiles 4w..4w+3.
// Fragment layout (ISA 7.12.2, 32-bit, wave32):
//   A (16x4):  lane half selects K{0,1}/{2,3}; lane&15 = M row; v2f = 2 K vals.
//   B (4x16):  lane half selects K{0,1}/{2,3}; lane&15 = N col; v2f = 2 K vals.
//   C/D: VGPR r -> M = r + 8*(lane>=16), N = lane&15. Real rows live in lanes 0-15.
// ---------------------------------------------------------------------------
__device__ __forceinline__ float mish_f(float x) {
  // stable softplus
  float sp = (x > 20.0f) ? x : log1pf(__expf(x));
  return x * tanhf(sp);
}
__device__ __forceinline__ float sigmoid_f(float x) {
  return 1.0f / (1.0f + __expf(-x));
}

__global__ __launch_bounds__(512) void se4_mlp_kernel(
    const float* __restrict__ pooled,
    const float* __restrict__ reduce_w, const float* __restrict__ reduce_b,
    const float* __restrict__ expand_w, const float* __restrict__ expand_b,
    float* __restrict__ gates) {
  __shared__ float S[8 * 256];  // layer-1 activations, rows 0..7 only

  int tid = threadIdx.x;
  int lane = tid & 31;
  int wv = tid >> 5;           // 0..15
  int nlo = lane & 15;         // N column within tile / M row for A
  int koff = (lane >> 4) << 1; // 0 or 2: this lane half's K offset
  int row = nlo;               // A-matrix M row carried by this lane
  float rmask = (row < 8) ? 1.0f : 0.0f;  // zero out pad rows (no divergence)
  int rsafe = row & 7;

  // ---- Layer 1 ----
  {
    int n0 = (wv << 4) + nlo;                        // 0..255
    const float* arow = pooled + rsafe * 1024;
    const float* brow = reduce_w + (size_t)n0 * 1024;
    v8f acc = {0.f, 0.f, 0.f, 0.f, 0.f, 0.f, 0.f, 0.f};
    for (int k = 0; k < 1024; k += 4) {
      v2f a, bm;
      a.x = arow[k + koff] * rmask;
      a.y = arow[k + koff + 1] * rmask;
      bm.x = brow[k + koff];
      bm.y = brow[k + koff + 1];
      acc = __builtin_amdgcn_wmma_f32_16x16x4_f32(
          false, a, false, bm, (short)0, acc, false, false);
    }
    if (lane < 16) {
      float bias = reduce_b[n0];
#pragma unroll
      for (int r = 0; r < 8; ++r)
        S[r * 256 + n0] = mish_f(acc[r] + bias);
    }
  }
  __syncthreads();

  // ---- Layer 2 ----
  for (int tI = 0; tI < 4; ++tI) {
    int n0 = (((wv << 2) + tI) << 4) + nlo;          // 0..1023
    const float* brow = expand_w + (size_t)n0 * 256;
    const float* srow = S + rsafe * 256;
    v8f acc = {0.f, 0.f, 0.f, 0.f, 0.f, 0.f, 0.f, 0.f};
    for (int k = 0; k < 256; k += 4) {
      v2f a, bm;
      a.x = srow[k + koff] * rmask;
      a.y = srow[k + koff + 1] * rmask;
      bm.x = brow[k + koff];
      bm.y = brow[k + koff + 1];
      acc = __builtin_amdgcn_wmma_f32_16x16x4_f32(
          false, a, false, bm, (short)0, acc, false, false);
    }
    if (lane < 16) {
      float bias = expand_b[n0];
#pragma unroll
      for (int r = 0; r < 8; ++r)
        gates[r * 1024 + n0] = sigmoid_f(acc[r] + bias);
    }
  }
}

// ---------------------------------------------------------------------------
// Kernel 3: apply gates. out = t * gate(b,c,i,j). float4 per thread.
// t read with NT (last-use) hint: already L2-resident from pass 1, never
// needed again. out stored NT so the write stream doesn't evict not-yet-read
// t lines from L2.
// ---------------------------------------------------------------------------
__global__ __launch_bounds__(256) void se4_apply_kernel(
    const float* __restrict__ t, const float* __restrict__ gates,
    float* __restrict__ out) {
  size_t f4 = (size_t)blockIdx.x * 256 + threadIdx.x;
  size_t e = f4 * 4;
  int b = (int)(e >> 22);
  int c = (int)((e >> 16) & 63);
  int h = (int)((e >> 8) & 255);
  int w = (int)(e & 255);
  int i = h >> 6;
  int j = w >> 6;
  float g = gates[b * 1024 + ((i * 4 + j) << 6) + c];

  v4f v = __builtin_nontemporal_load((const v4f*)(t + e));
  v.x *= g; v.y *= g; v.z *= g; v.w *= g;
  __builtin_nontemporal_store(v, (v4f*)(out + e));
}

// ---------------------------------------------------------------------------
extern "C" void kernel_launch(void* const* d_in, const int* in_sizes, int n_in,
                              void* d_out, int out_size, void* d_ws,
                              size_t ws_size, hipStream_t stream) {
  const float* t        = (const float*)d_in[0];
  const float* reduce_w = (const float*)d_in[1];
  const float* reduce_b = (const float*)d_in[2];
  const float* expand_w = (const float*)d_in[3];
  const float* expand_b = (const float*)d_in[4];
  float* out = (float*)d_out;

  float* pooled = (float*)d_ws;          // 8*1024 f32 = 32 KB
  float* gates  = pooled + 8 * 1024;     // 8*1024 f32 = 32 KB

  se4_pool_kernel<<<8192, 256, 0, stream>>>(t, pooled);
  se4_mlp_kernel<<<1, 512, 0, stream>>>(pooled, reduce_w, reduce_b,
                                        expand_w, expand_b, gates);
  se4_apply_kernel<<<32768, 256, 0, stream>>>(t, gates, out);
}